// Attention_3083786519278
// MI455X (gfx1250) — compile-verified
//
#include <hip/hip_runtime.h>
#include <hip/hip_bf16.h>

typedef __attribute__((ext_vector_type(2))) float v2f;
typedef __attribute__((ext_vector_type(8))) float v8f;

#define BATCH 1024
#define DIM   1024
#define HEADS 16
#define NB    128
#define DX    100
#define DXP   112          // DX padded to multiple of 16 for WMMA N
#define INV_SQRT_2PI 0.3989422804014327f

// ---------------------------------------------------------------------------
// alpha = x @ W_alpha^T + b_alpha        [1024,1024] = [1024,1024]x[1024,1024]^T
// NT GEMM via V_WMMA_F32_16X16X4_F32. Each wave: 16x32 output, K=1024.
// ---------------------------------------------------------------------------
__global__ __launch_bounds__(256) void alpha_gemm(const float* __restrict__ x,
                                                  const float* __restrict__ Wa,
                                                  const float* __restrict__ ba,
                                                  float* __restrict__ alpha) {
    const int tid    = threadIdx.x;
    const int lane   = tid & 31;
    const int laneLo = lane & 15;
    const int laneHi = lane >> 4;
    const int wid    = blockIdx.x * 8 + (tid >> 5);   // 0..2047
    const int mTile  = wid >> 5;                      // 0..63
    const int nPair  = wid & 31;                      // 0..31
    const int m0     = mTile * 16;
    const int n0     = nPair * 32;

    const float* ap  = x  + (size_t)(m0 + laneLo) * DIM + 2 * laneHi;
    const float* bp0 = Wa + (size_t)(n0 + laneLo) * DIM + 2 * laneHi;
    const float* bp1 = bp0 + 16 * DIM;

    v8f acc0 = {};
    v8f acc1 = {};
#pragma unroll 4
    for (int k = 0; k < DIM; k += 4) {
        v2f a  = *(const v2f*)ap;
        v2f b0 = *(const v2f*)bp0;
        v2f b1 = *(const v2f*)bp1;
        acc0 = __builtin_amdgcn_wmma_f32_16x16x4_f32(false, a, false, b0,
                                                     (short)0, acc0, false, false);
        acc1 = __builtin_amdgcn_wmma_f32_16x16x4_f32(false, a, false, b1,
                                                     (short)0, acc1, false, false);
        ap += 4; bp0 += 4; bp1 += 4;
    }

    const float bias0 = ba[n0 + laneLo];
    const float bias1 = ba[n0 + 16 + laneLo];
#pragma unroll
    for (int r = 0; r < 8; ++r) {
        const int row = m0 + r + 8 * laneHi;
        alpha[(size_t)row * 1024 + n0 + laneLo]      = acc0[r] + bias0;
        alpha[(size_t)row * 1024 + n0 + 16 + laneLo] = acc1[r] + bias1;
    }
}

// ---------------------------------------------------------------------------
// mu = x@W_mu^T + b_mu ; sigma_sq = softplus(x@W_sig^T + b_sig)   [1024,16]
// One A-tile feeds two accumulators (shared K loop).
// ---------------------------------------------------------------------------
__global__ __launch_bounds__(256) void musig_gemm(const float* __restrict__ x,
                                                  const float* __restrict__ Wmu,
                                                  const float* __restrict__ bmu,
                                                  const float* __restrict__ Wsig,
                                                  const float* __restrict__ bsig,
                                                  float* __restrict__ muOut,
                                                  float* __restrict__ sigOut) {
    const int tid    = threadIdx.x;
    const int lane   = tid & 31;
    const int laneLo = lane & 15;
    const int laneHi = lane >> 4;
    const int wid    = blockIdx.x * 8 + (tid >> 5);   // 0..63 (mTile)
    const int m0     = wid * 16;

    const float* ap = x    + (size_t)(m0 + laneLo) * DIM + 2 * laneHi;
    const float* bm = Wmu  + (size_t)laneLo * DIM + 2 * laneHi;
    const float* bs = Wsig + (size_t)laneLo * DIM + 2 * laneHi;

    v8f accM = {};
    v8f accS = {};
#pragma unroll 4
    for (int k = 0; k < DIM; k += 4) {
        v2f a   = *(const v2f*)ap;
        v2f bmv = *(const v2f*)bm;
        v2f bsv = *(const v2f*)bs;
        accM = __builtin_amdgcn_wmma_f32_16x16x4_f32(false, a, false, bmv,
                                                     (short)0, accM, false, false);
        accS = __builtin_amdgcn_wmma_f32_16x16x4_f32(false, a, false, bsv,
                                                     (short)0, accS, false, false);
        ap += 4; bm += 4; bs += 4;
    }

    const float biasM = bmu[laneLo];
    const float biasS = bsig[laneLo];
#pragma unroll
    for (int r = 0; r < 8; ++r) {
        const int row = m0 + r + 8 * laneHi;
        const int h   = laneLo;
        const float m = accM[r] + biasM;
        const float z = accS[r] + biasS;
        // numerically stable softplus
        const float sp = fmaxf(z, 0.0f) + log1pf(expf(-fabsf(z)));
        muOut[row * HEADS + h]  = m;
        sigOut[row * HEADS + h] = sp;
    }
}

// ---------------------------------------------------------------------------
// phi1w^T [DXP][NB]: basis pdfs on quadrature grid, trapezoid weight folded in,
// rows d >= DX zero-padded.
// ---------------------------------------------------------------------------
__global__ void phi1w_kernel(float* __restrict__ phiT) {
    const int idx = blockIdx.x * blockDim.x + threadIdx.x;
    if (idx >= DXP * NB) return;
    const int d = idx / NB;
    const int n = idx % NB;
    float val = 0.0f;
    if (d < DX) {
        const float mu_b = (float)(n >> 1) * (1.0f / 63.0f);   // repeat(linspace(0,1,64),2)
        const float sig  = (n & 1) ? 0.5f : 0.1f;              // tile([0.1,0.5],64)
        const float T    = (float)d * (1.0f / 99.0f);
        const float t    = (mu_b - T) / sig;
        const float h    = 1.0f / 99.0f;
        const float w    = (d == 0 || d == DX - 1) ? 0.5f * h : h;
        val = expf(-0.5f * t * t) * (INV_SQRT_2PI / sig) * w;
    }
    phiT[idx] = val;
}

// ---------------------------------------------------------------------------
// g = B @ phi1w^T   [1024,112] = [1024,128] x [112,128]^T  (NT, WMMA f32)
// ---------------------------------------------------------------------------
__global__ __launch_bounds__(256) void g_gemm(const float* __restrict__ Bin,
                                              const float* __restrict__ phiT,
                                              float* __restrict__ g) {
    const int tid    = threadIdx.x;
    const int lane   = tid & 31;
    const int laneLo = lane & 15;
    const int laneHi = lane >> 4;
    const int wid    = blockIdx.x * 8 + (tid >> 5);   // 0..447
    const int mTile  = wid / 7;                       // 0..63
    const int nTile  = wid % 7;                       // 0..6
    const int m0     = mTile * 16;
    const int n0     = nTile * 16;

    const float* ap = Bin  + (size_t)(m0 + laneLo) * NB + 2 * laneHi;
    const float* bp = phiT + (size_t)(n0 + laneLo) * NB + 2 * laneHi;

    v8f acc = {};
#pragma unroll 4
    for (int k = 0; k < NB; k += 4) {
        v2f a = *(const v2f*)ap;
        v2f b = *(const v2f*)bp;
        acc = __builtin_amdgcn_wmma_f32_16x16x4_f32(false, a, false, b,
                                                    (short)0, acc, false, false);
        ap += 4; bp += 4;
    }
#pragma unroll
    for (int r = 0; r < 8; ++r) {
        const int row = m0 + r + 8 * laneHi;
        g[(size_t)row * DXP + n0 + laneLo] = acc[r];
    }
}

// ---------------------------------------------------------------------------
// c[b,h] = sum_d N(mu,sigma)(T_d) * g[b,d]    (w already folded into g)
// One wave32 per (row, head); shfl reduction.
// ---------------------------------------------------------------------------
__global__ __launch_bounds__(512) void final_kernel(const float* __restrict__ muOut,
                                                    const float* __restrict__ sigOut,
                                                    const float* __restrict__ g,
                                                    float* __restrict__ cOut) {
    const int b    = blockIdx.x;
    const int h    = threadIdx.x >> 5;   // 16 waves -> 16 heads
    const int lane = threadIdx.x & 31;

    const float mu      = muOut[b * HEADS + h];
    const float ss      = sigOut[b * HEADS + h];
    const float inv_std = rsqrtf(ss);

    float s = 0.0f;
#pragma unroll
    for (int i = 0; i < 4; ++i) {
        const int d = lane + 32 * i;
        if (d < DX) {
            const float T   = (float)d * (1.0f / 99.0f);
            const float t   = (mu - T) * inv_std;
            const float pdf = expf(-0.5f * t * t) * INV_SQRT_2PI * inv_std;
            s += pdf * g[(size_t)b * DXP + d];
        }
    }
#pragma unroll
    for (int off = 16; off > 0; off >>= 1)
        s += __shfl_xor(s, off, 32);
    if (lane == 0) cOut[b * HEADS + h] = s;
}

// ---------------------------------------------------------------------------
extern "C" void kernel_launch(void* const* d_in, const int* in_sizes, int n_in,
                              void* d_out, int out_size, void* d_ws, size_t ws_size,
                              hipStream_t stream) {
    (void)in_sizes; (void)n_in; (void)out_size; (void)ws_size;

    const float* x    = (const float*)d_in[0];
    const float* Bin  = (const float*)d_in[1];
    const float* Wmu  = (const float*)d_in[2];
    const float* bmu  = (const float*)d_in[3];
    const float* Wsig = (const float*)d_in[4];
    const float* bsig = (const float*)d_in[5];
    const float* Wa   = (const float*)d_in[6];
    const float* ba   = (const float*)d_in[7];

    float* out      = (float*)d_out;
    float* cOut     = out;                           // [1024,16]
    float* muOut    = out + BATCH * HEADS;           // [1024,16]
    float* sigOut   = out + 2 * BATCH * HEADS;       // [1024,16]
    float* alphaOut = out + 3 * BATCH * HEADS;       // [1024,1024]

    float* phiT = (float*)d_ws;                      // [112][128]
    float* g    = phiT + DXP * NB;                   // [1024][112]

    alpha_gemm<<<256, 256, 0, stream>>>(x, Wa, ba, alphaOut);
    musig_gemm<<<8, 256, 0, stream>>>(x, Wmu, bmu, Wsig, bsig, muOut, sigOut);
    phi1w_kernel<<<(DXP * NB + 255) / 256, 256, 0, stream>>>(phiT);
    g_gemm<<<56, 256, 0, stream>>>(Bin, phiT, g);
    final_kernel<<<BATCH, 512, 0, stream>>>(muOut, sigOut, g, cOut);
}